// TopKSpatialFusion_11802570130086
// MI455X (gfx1250) — compile-verified
//
#include <hip/hip_runtime.h>
#include <hip/hip_bf16.h>

#define K_TOP    30
#define BB       32
#define CC       384
#define HH       56
#define WW       56
#define HW       (HH * WW)      // 3136
#define XS_STRIDE 64            // floats per LDS row (256B, keeps B128 dsts 16B aligned)
#define XS_ROWS   58            // 56 + halo
#define XS_COL0   4             // interior starts at col 4 -> 16B-aligned rows
#define NT        256

#define AS1 __attribute__((address_space(1)))
#define AS3 __attribute__((address_space(3)))

typedef int   v4i __attribute__((vector_size(16)));
typedef float v2f __attribute__((ext_vector_type(2)));
typedef float v8f __attribute__((ext_vector_type(8)));

#if __has_builtin(__builtin_amdgcn_global_load_async_to_lds_b128)
#define HAVE_ASYNC_LOAD 1
#else
#define HAVE_ASYNC_LOAD 0
#endif
#if __has_builtin(__builtin_amdgcn_global_store_async_from_lds_b128)
#define HAVE_ASYNC_STORE 1
#else
#define HAVE_ASYNC_STORE 0
#endif
#if __has_builtin(__builtin_amdgcn_wmma_f32_16x16x4_f32)
#define HAVE_WMMA_SCAN 1
#else
#define HAVE_WMMA_SCAN 0
#endif

__device__ __forceinline__ void wait_async0() {
#if __has_builtin(__builtin_amdgcn_s_wait_asynccnt)
  __builtin_amdgcn_s_wait_asynccnt(0);
#elif HAVE_ASYNC_LOAD || HAVE_ASYNC_STORE
  asm volatile("s_wait_asynccnt 0" ::: "memory");
#endif
}

__device__ __forceinline__ unsigned ballot32(bool p) {
#if __has_builtin(__builtin_amdgcn_ballot_w32)
  return __builtin_amdgcn_ballot_w32(p);
#else
  return (unsigned)__ballot(p);
#endif
}

__global__ __launch_bounds__(NT)
void topk_spatial_kernel(const float* __restrict__ x,
                         const float* __restrict__ weight,
                         const float* __restrict__ bias,
                         float* __restrict__ out) {
  __shared__ __align__(16) float    xs[XS_ROWS * XS_STRIDE];  // 14848 B
  __shared__ __align__(16) unsigned keys[HW];                 // 12544 B
  __shared__ __align__(16) unsigned hist[256];                // histogram, then column-suffix
  __shared__ unsigned selInfo[2];   // [0]=selected byte, [1]=remaining K
  __shared__ unsigned wTot[8];
  __shared__ unsigned st[16];       // cross-column suffix of column totals

  const int t   = threadIdx.x;
  const int img = blockIdx.x;                    // b*CC + c
  const int c   = img % CC;
  const long long base = (long long)img * HW;
  const float* src = x + base;
  float*       dst = out + base;

  // ---- zero LDS tile (provides SAME-padding halo), B128 stores ----
  {
    const float4 z4 = make_float4(0.f, 0.f, 0.f, 0.f);
    float4* xs4 = (float4*)xs;
    for (int i = t; i < (XS_ROWS * XS_STRIDE) / 4; i += NT) xs4[i] = z4;
  }
  __syncthreads();

  // ---- stage 56x56 interior into LDS (async B128 when available) ----
#if HAVE_ASYNC_LOAD
  for (int i = t; i < HW / 4; i += NT) {                 // 784 x B128
    const int r = i / (WW / 4), q = i % (WW / 4);
    const float* g = src + r * WW + q * 4;
    float*       l = &xs[(r + 1) * XS_STRIDE + XS_COL0 + q * 4];
    __builtin_amdgcn_global_load_async_to_lds_b128((AS1 v4i*)g, (AS3 v4i*)l, 0, 0);
  }
  wait_async0();
#else
  for (int i = t; i < HW / 4; i += NT) {
    const int r = i / (WW / 4), q = i % (WW / 4);
    const float4 v = *(const float4*)(src + r * WW + q * 4);
    *(float4*)&xs[(r + 1) * XS_STRIDE + XS_COL0 + q * 4] = v;
  }
#endif
  __syncthreads();

  // ---- depthwise 3x3 conv -> order-preserving uint keys in LDS ----
  const float* wp = weight + c * 9;   // c is block-uniform -> scalar loads
  const float w00 = wp[0], w01 = wp[1], w02 = wp[2];
  const float w10 = wp[3], w11 = wp[4], w12 = wp[5];
  const float w20 = wp[6], w21 = wp[7], w22 = wp[8];
  const float bv  = bias[c];

  for (int p = t; p < HW; p += NT) {
    const int h = p / WW, w = p % WW;
    // output (h,w) taps XS(h+dh, w+3+dw), dh,dw in 0..2 (halo already zero)
    const float* rr = &xs[h * XS_STRIDE + w + 3];
    float s = bv;
    s += w00 * rr[0] + w01 * rr[1] + w02 * rr[2];  rr += XS_STRIDE;
    s += w10 * rr[0] + w11 * rr[1] + w12 * rr[2];  rr += XS_STRIDE;
    s += w20 * rr[0] + w21 * rr[1] + w22 * rr[2];
    const unsigned u = __float_as_uint(s);
    keys[p] = (u & 0x80000000u) ? ~u : (u | 0x80000000u);   // monotone mapping
  }
  __syncthreads();

  // ---- exact K-th largest via 4x 8-bit LDS radix select ----
  // Bin b: column n = b>>4, row m = b&15.  H[k][n] = hist[n*16+k].
  unsigned Krem = K_TOP;
  unsigned pref = 0;
#pragma unroll
  for (int pass = 0; pass < 4; ++pass) {
    const int sh = 24 - pass * 8;
    hist[t] = 0u;
    __syncthreads();
    for (int p = t; p < HW; p += NT) {
      const unsigned k = keys[p];
      bool act;
      if (pass == 0) act = true;
      else           act = ((k >> (sh + 8)) == pref);
      if (act) atomicAdd(&hist[(k >> sh) & 0xFFu], 1u);
    }
    __syncthreads();
    const unsigned mycount = hist[t];
    __syncthreads();   // all mycount reads done before hist is overwritten/rescanned

#if HAVE_WMMA_SCAN
    // Suffix-scan the 16x16 histogram on the matrix pipe (one wave):
    //   ColSuffix(16x16) = Tri(16x16) @ H(16x16),  Tri[m,k] = (k >= m)
    // chained as 4x V_WMMA_F32_16X16X4_F32 (exact: counts < 2^24).
    if (t < 32) {
      const int mn  = t & 15;            // M for A-operand, N for B-operand
      const int kk0 = (t >> 4) << 1;     // K lane-split: VGPR0 holds K=kk0, VGPR1 holds K=kk0+1
      v8f acc = {};
#pragma unroll
      for (int j = 0; j < 4; ++j) {
        const int k0 = 4 * j + kk0;
        v2f a, b;
        a.x = (k0     >= mn) ? 1.0f : 0.0f;     // triangular constant (documented A layout)
        a.y = (k0 + 1 >= mn) ? 1.0f : 0.0f;
        b.x = (float)hist[mn * 16 + k0];        // H column mn, rows k0, k0+1
        b.y = (float)hist[mn * 16 + k0 + 1];
        acc = __builtin_amdgcn_wmma_f32_16x16x4_f32(false, a, false, b,
                                                    (short)0, acc, false, false);
      }
      // D layout: VGPR g, lane t -> n = t&15, m = g + 8*(t>>4); write colsuffix in place
#pragma unroll
      for (int g = 0; g < 8; ++g) {
        const int m = g + ((t >> 4) << 3);
        hist[mn * 16 + m] = (unsigned)(acc[g] + 0.5f);
      }
    }
    __syncthreads();
    if (t < 16) {  // cross-column suffix of column totals T[n] = ColSuffix[0][n]
      unsigned s = 0;
      for (int n2 = 15; n2 >= t; --n2) s += hist[n2 * 16];
      st[t] = s;
    }
    __syncthreads();
    const unsigned S = hist[t] + (((t >> 4) < 15) ? st[(t >> 4) + 1] : 0u);
#else
    // Fallback: Hillis-Steele inclusive suffix sum over 256 bins
    for (int d = 1; d < 256; d <<= 1) {
      const unsigned other = (t + d < 256) ? hist[t + d] : 0u;
      __syncthreads();
      hist[t] += other;
      __syncthreads();
    }
    const unsigned S = hist[t];
#endif
    const unsigned g = S - mycount;      // count of keys strictly greater than bin t
    if (S >= Krem && g < Krem) { selInfo[0] = (unsigned)t; selInfo[1] = Krem - g; }
    __syncthreads();
    pref = (pref << 8) | selInfo[0];
    Krem = selInfo[1];
    __syncthreads();
  }
  const unsigned Tkey   = pref;   // exact 30th-largest key
  const unsigned r_ties = Krem;   // threshold-equal keys to keep (lowest index first)

  // ---- stable tie ranking (wave32 ballot) + masked output ----
  const unsigned lane = (unsigned)t & 31u;
  const unsigned wv   = (unsigned)t >> 5;
  unsigned tieBase = 0;
  for (int chunk = 0; chunk < 13; ++chunk) {
    const int  p     = chunk * NT + t;
    const bool valid = (p < HW);
    const unsigned k = valid ? keys[p] : 0u;
    const bool eq    = valid && (k == Tkey);
    const unsigned m = ballot32(eq);
    if (lane == 0) wTot[wv] = (unsigned)__popc(m);
    __syncthreads();
    unsigned baseW = 0, tot = 0;
#pragma unroll
    for (int j = 0; j < 8; ++j) { const unsigned v = wTot[j]; tot += v; if (j < (int)wv) baseW += v; }
    const unsigned rank = tieBase + baseW + (unsigned)__popc(m & ((1u << lane) - 1u));
    const bool keep = (k > Tkey) || (eq && rank < r_ties);
    if (valid) {
      const int h = p / WW, w = p % WW;
      const float xv  = xs[(h + 1) * XS_STRIDE + (w + XS_COL0)];
      const float val = keep ? xv : 0.0f;
#if HAVE_ASYNC_STORE
      keys[p] = __float_as_uint(val);        // repack result tile in LDS
#else
      dst[p] = val;
#endif
    }
    tieBase += tot;
    __syncthreads();
  }

#if HAVE_ASYNC_STORE
  for (int i = t; i < HW / 4; i += NT) {      // 784 x B128 LDS -> global
    float*    g = dst + i * 4;
    unsigned* l = &keys[i * 4];
    __builtin_amdgcn_global_store_async_from_lds_b128((AS1 v4i*)g, (AS3 v4i*)l, 0, 0);
  }
  wait_async0();
#endif
}

extern "C" void kernel_launch(void* const* d_in, const int* in_sizes, int n_in,
                              void* d_out, int out_size, void* d_ws, size_t ws_size,
                              hipStream_t stream) {
  (void)in_sizes; (void)n_in; (void)out_size; (void)d_ws; (void)ws_size;
  const float* x      = (const float*)d_in[0];
  const float* weight = (const float*)d_in[1];
  const float* bias   = (const float*)d_in[2];
  float* out          = (float*)d_out;
  dim3 grid(BB * CC);
  dim3 block(NT);
  topk_spatial_kernel<<<grid, block, 0, stream>>>(x, weight, bias, out);
}